// CorrelationVolume_51599737094156
// MI455X (gfx1250) — compile-verified
//
#include <hip/hip_runtime.h>

typedef __attribute__((ext_vector_type(2))) float v2f;
typedef __attribute__((ext_vector_type(8))) float v8f;

#define B_DIM   4
#define C_DIM   256
#define H_DIM   128
#define W_DIM   256
#define D_DISP  48
#define HW      (H_DIM * W_DIM)

#define WCHUNK  128
#define LCOLS   128
#define RCOLS   176
// channel-pair interleaved layouts: element (c, col) -> [(c>>1)*STRIDE + 2*col + (c&1)]
#define L2STRIDE 288    // 2*128 + 32 pad ; 288 % 64 == 32 -> lane halves on disjoint banks
#define R2STRIDE 352    // 2*176          ; 352 % 64 == 32 -> lane halves on disjoint banks
#define NPAIR    (C_DIM / 2)   // 128 pair-rows

// norm scale arrays hidden in L's pad region (cols 256..287 of pair-rows 0..9)
#define INVL(w) Ls2[((w) >> 5) * L2STRIDE + 2 * LCOLS + ((w) & 31)]
#define INVR(j) Ls2[(4 + ((j) >> 5)) * L2STRIDE + 2 * LCOLS + ((j) & 31)]

#define SMEM_FLOATS (L2STRIDE * NPAIR + R2STRIDE * NPAIR)   // 81,920 -> 320 KB exactly

__launch_bounds__(256, 1)
__global__ void corrvol_wmma_f32(const float* __restrict__ Lg,
                                 const float* __restrict__ Rg,
                                 float* __restrict__ out) {
    extern __shared__ float smem[];
    float* Ls2 = smem;                        // [NPAIR][L2STRIDE]
    float* Rs2 = smem + L2STRIDE * NPAIR;     // [NPAIR][R2STRIDE]

    const int tid  = threadIdx.x;
    const int wg   = blockIdx.x;
    const int wcId = wg & 1;                  // which 128-wide w chunk
    const int h    = (wg >> 1) & (H_DIM - 1);
    const int b    = wg >> 8;
    const int wc   = wcId * WCHUNK;

    // ---------- phase 1: stage raw L / R chunks into LDS (pair-interleaved) ----------
    {
        const int w    = tid & 127;
        const int csub = tid >> 7;            // 0..1 -> c = 2*it + csub
        const float* Lbase = Lg + ((b * C_DIM) * H_DIM + h) * W_DIM + wc + w;
#pragma unroll 4
        for (int it = 0; it < 128; ++it) {
            // (c>>1)=it, (c&1)=csub
            Ls2[it * L2STRIDE + 2 * w + csub] = Lbase[(it * 2 + csub) * HW];
        }
    }
    {
        const float* Rbase = Rg + ((b * C_DIM) * H_DIM + h) * W_DIM;
#pragma unroll 2
        for (int it = 0; it < 176; ++it) {
            int idx = it * 256 + tid;         // 0 .. 45055
            int c   = idx / RCOLS;
            int j   = idx - c * RCOLS;        // 0..175
            int wp  = wc - 48 + j;            // w' in [-48, wc+127]
            float v = 0.0f;
            if (wp >= 0) v = Rbase[c * HW + wp];
            Rs2[(c >> 1) * R2STRIDE + 2 * j + (c & 1)] = v;  // zero-fill == reference pad
        }
    }
    __syncthreads();

    // ---------- phase 2: per-column L2 norms (304 columns, 256 threads) ----------
    for (int col = tid; col < LCOLS + RCOLS; col += 256) {
        float s = 0.0f;
        if (col < LCOLS) {
            for (int c = 0; c < C_DIM; ++c) {
                float x = Ls2[(c >> 1) * L2STRIDE + 2 * col + (c & 1)];
                s += x * x;
            }
            INVL(col) = 1.0f / fmaxf(sqrtf(s), 1e-12f);
        } else {
            int j = col - LCOLS;
            for (int c = 0; c < C_DIM; ++c) {
                float x = Rs2[(c >> 1) * R2STRIDE + 2 * j + (c & 1)];
                s += x * x;
            }
            INVR(j) = 1.0f / fmaxf(sqrtf(s), 1e-12f);
        }
    }
    __syncthreads();

    // ---------- phase 3: scale in place (row-parallel) ----------
    {
        const int w    = tid & 127;
        const int csub = tid >> 7;
        float sc = INVL(w);
#pragma unroll 4
        for (int it = 0; it < 128; ++it) {
            Ls2[it * L2STRIDE + 2 * w + csub] *= sc;
        }
    }
    {
#pragma unroll 2
        for (int it = 0; it < 176; ++it) {
            int idx = it * 256 + tid;
            int c   = idx / RCOLS;
            int j   = idx - c * RCOLS;
            Rs2[(c >> 1) * R2STRIDE + 2 * j + (c & 1)] *= INVR(j);
        }
    }
    __syncthreads();

    // ---------- phase 4: banded GEMM via V_WMMA_F32_16X16X4_F32 ----------
    // wave = w-tile (0..7); each wave owns all 4 w'-tiles of its band slice.
    const int wt   = tid >> 5;
    const int lane = tid & 31;
    const int half = lane >> 4;               // K-half selector (pair-row +half)
    const int m    = lane & 15;               // M (A rows) / N (B cols)

    // K-step kk: A lane pair = L[{4kk+2h, 4kk+2h+1}][wt*16+m] = pair-row (2kk+h)
    const float* Ap = Ls2 + half * L2STRIDE + 2 * (wt * 16 + m);
    const float* Bp = Rs2 + half * R2STRIDE + 2 * (wt * 16 + m);

    v8f acc0 = {}, acc1 = {}, acc2 = {}, acc3 = {};
#pragma unroll 4
    for (int kk = 0; kk < 64; ++kk) {
        v2f a  = *(const v2f*)(Ap);
        v2f b0 = *(const v2f*)(Bp);
        v2f b1 = *(const v2f*)(Bp + 32);      // +16 cols
        v2f b2 = *(const v2f*)(Bp + 64);
        v2f b3 = *(const v2f*)(Bp + 96);
        acc0 = __builtin_amdgcn_wmma_f32_16x16x4_f32(false, a, false, b0, (short)0, acc0, false, false);
        acc1 = __builtin_amdgcn_wmma_f32_16x16x4_f32(false, a, false, b1, (short)0, acc1, false, false);
        acc2 = __builtin_amdgcn_wmma_f32_16x16x4_f32(false, a, false, b2, (short)0, acc2, false, false);
        acc3 = __builtin_amdgcn_wmma_f32_16x16x4_f32(false, a, false, b3, (short)0, acc3, false, false);
        Ap += 2 * L2STRIDE;                   // advance 4 channels
        Bp += 2 * R2STRIDE;
    }

    // ---------- phase 5: scatter stores  out[b, i=w-w', h, w] ----------
    const int gw_base = wc + wt * 16;
    const int wp0     = wc - 48 + wt * 16 + m;      // w' for delta=0; delta adds 16
    float* outB = out + (b * D_DISP * H_DIM) * W_DIM + h * W_DIM;
#pragma unroll
    for (int r = 0; r < 8; ++r) {
        int w = gw_base + r + 8 * half;             // C/D layout: M = r + 8*half
        int i;
        i = w - wp0;
        if (i >= 0 && i < D_DISP) outB[i * HW + w] = acc0[r];
        i = w - (wp0 + 16);
        if (i >= 0 && i < D_DISP) outB[i * HW + w] = acc1[r];
        i = w - (wp0 + 32);
        if (i >= 0 && i < D_DISP) outB[i * HW + w] = acc2[r];
        i = w - (wp0 + 48);
        if (i >= 0 && i < D_DISP) outB[i * HW + w] = acc3[r];
    }
}

extern "C" void kernel_launch(void* const* d_in, const int* in_sizes, int n_in,
                              void* d_out, int out_size, void* d_ws, size_t ws_size,
                              hipStream_t stream) {
    (void)in_sizes; (void)n_in; (void)out_size; (void)d_ws; (void)ws_size;
    const float* L = (const float*)d_in[0];
    const float* R = (const float*)d_in[1];
    float* O = (float*)d_out;

    const size_t smem_bytes = (size_t)SMEM_FLOATS * sizeof(float); // 327,680 B == 320 KB
    static_assert(SMEM_FLOATS * sizeof(float) <= 320u * 1024u, "LDS budget");

    hipFuncSetAttribute(reinterpret_cast<const void*>(corrvol_wmma_f32),
                        hipFuncAttributeMaxDynamicSharedMemorySize,
                        (int)smem_bytes);

    dim3 grid(B_DIM * H_DIM * (W_DIM / WCHUNK));   // 1024
    dim3 block(256);
    corrvol_wmma_f32<<<grid, block, smem_bytes, stream>>>(L, R, O);
}